// Loss_39170101740026
// MI455X (gfx1250) — compile-verified
//
#include <hip/hip_runtime.h>
#include <hip/hip_bf16.h>
#include <math.h>

// ---- problem constants (fixed by reference setup_inputs) ----
#define BATCH   8
#define N_COARSE 1024
#define N_FINE   8192
#define N_GT     8192
#define TILES_C  (N_COARSE / 16)   // 64
#define TILES_F  (N_FINE / 16)     // 512
#define TILES_G  (N_GT / 16)       // 512

typedef __attribute__((ext_vector_type(2))) float v2f;
typedef __attribute__((ext_vector_type(8))) float v8f;

// ---------------------------------------------------------------------------
// Prep: pack points as (p0, p1, p2, ||p||^2) for x-sets (row side), and
// (-2p0, -2p1, -2p2, ||p||^2) for the gt set (column side, channel-first).
// ---------------------------------------------------------------------------
__global__ void prep_x_kernel(const float* __restrict__ pts, float* __restrict__ X4,
                              int total) {
    int i = blockIdx.x * blockDim.x + threadIdx.x;
    if (i >= total) return;
    float x = pts[3 * i + 0];
    float y = pts[3 * i + 1];
    float z = pts[3 * i + 2];
    X4[4 * i + 0] = x;
    X4[4 * i + 1] = y;
    X4[4 * i + 2] = z;
    X4[4 * i + 3] = x * x + y * y + z * z;
}

__global__ void prep_y_kernel(const float* __restrict__ gt, float* __restrict__ Y4,
                              int M, int total) {
    int i = blockIdx.x * blockDim.x + threadIdx.x;
    if (i >= total) return;
    int b = i / M;
    int m = i - b * M;
    const float* g = gt + (size_t)b * 3 * M;
    float x = g[m];
    float y = g[M + m];
    float z = g[2 * M + m];
    Y4[4 * i + 0] = -2.0f * x;
    Y4[4 * i + 1] = -2.0f * y;
    Y4[4 * i + 2] = -2.0f * z;
    Y4[4 * i + 3] = x * x + y * y + z * z;
}

// ---------------------------------------------------------------------------
// Row kernel: one wave owns one 16-row x-tile, loops over all y-tiles.
// A (fixed) = (x0,x1,x2,1); B (stream) = (-2y0,-2y1,-2y2,||y||^2).
// D = -2 x.y + ||y_j||^2 ; add ||x_i||^2 per row; running min per row.
// Output: per-wave sum of the 16 row-mins.
// ---------------------------------------------------------------------------
__global__ void chamfer_rows_kernel(const float* __restrict__ X4,
                                    const float* __restrict__ Y4,
                                    int tilesX, int tilesY, int nx, int ny,
                                    float* __restrict__ partial) {
    const int wave = (int)((blockIdx.x * blockDim.x + threadIdx.x) >> 5);
    const int lane = (int)(threadIdx.x & 31u);
    const int b    = wave / tilesX;
    const int xt   = wave - b * tilesX;
    const float* Xb = X4 + (size_t)b * nx * 4;
    const float* Yb = Y4 + (size_t)b * ny * 4;

    const int col = lane & 15;     // point-within-tile this lane serves
    const int hi  = lane >> 4;     // 0: comps(0,1)  1: comps(2,3)

    // fixed A tile (slot 3 -> 1.0)
    const float* ap = Xb + ((size_t)(xt * 16 + col) * 4 + 2 * hi);
    v2f a;
    a.x = ap[0];
    a.y = hi ? 1.0f : ap[1];

    // per-row norms for rows v + 8*hi  (8 lane-uniform-per-half loads)
    float xn[8];
#pragma unroll
    for (int v = 0; v < 8; ++v)
        xn[v] = Xb[(size_t)(xt * 16 + v + 8 * hi) * 4 + 3];

    float runMin[8];
#pragma unroll
    for (int v = 0; v < 8; ++v) runMin[v] = __builtin_inff();

    const v8f zero = {0.f, 0.f, 0.f, 0.f, 0.f, 0.f, 0.f, 0.f};
    for (int t = 0; t < tilesY; ++t) {
        const float* bp = Yb + ((size_t)(t * 16 + col) * 4 + 2 * hi);
        v2f bb;
        bb.x = bp[0];
        bb.y = bp[1];
        v8f acc = __builtin_amdgcn_wmma_f32_16x16x4_f32(
            false, a, false, bb, (short)0, zero, false, false);
#pragma unroll
        for (int v = 0; v < 8; ++v) {
            float d = acc[v] + xn[v];
            runMin[v] = fminf(runMin[v], d);
        }
    }

    // min across the 16 lanes of each half (columns), then sum the row-mins
    float s = 0.0f;
#pragma unroll
    for (int v = 0; v < 8; ++v) {
        float m = runMin[v];
        m = fminf(m, __shfl_xor(m, 1, 32));
        m = fminf(m, __shfl_xor(m, 2, 32));
        m = fminf(m, __shfl_xor(m, 4, 32));
        m = fminf(m, __shfl_xor(m, 8, 32));
        s += m;
    }
    s += __shfl_xor(s, 16, 32);   // rows 0-7 (low half) + rows 8-15 (high half)
    if (lane == 0) partial[wave] = s;
}

// ---------------------------------------------------------------------------
// Column kernel: one wave owns one 16-col y-tile, loops over all x-tiles.
// B (fixed) = (-2y0,-2y1,-2y2,1); A (stream) = (x0,x1,x2,||x||^2).
// D = -2 x.y + ||x_i||^2 ; add ||y_j||^2 per lane; running min per column.
// Output: per-wave sum of the 16 column-mins.
// ---------------------------------------------------------------------------
__global__ void chamfer_cols_kernel(const float* __restrict__ X4,
                                    const float* __restrict__ Y4,
                                    int tilesX, int tilesY, int nx, int ny,
                                    float* __restrict__ partial) {
    const int wave = (int)((blockIdx.x * blockDim.x + threadIdx.x) >> 5);
    const int lane = (int)(threadIdx.x & 31u);
    const int b    = wave / tilesY;
    const int yt   = wave - b * tilesY;
    const float* Xb = X4 + (size_t)b * nx * 4;
    const float* Yb = Y4 + (size_t)b * ny * 4;

    const int col = lane & 15;
    const int hi  = lane >> 4;

    // fixed B tile
    const float* bp = Yb + ((size_t)(yt * 16 + col) * 4 + 2 * hi);
    v2f bb;
    bb.x = bp[0];
    bb.y = bp[1];
    // ||y_j||^2 lives in high-half lanes' .y — broadcast to the low half
    float ynPartner = __shfl_xor(bb.y, 16, 32);
    float yn = hi ? bb.y : ynPartner;
    if (hi) bb.y = 1.0f;

    float runCol = __builtin_inff();
    const v8f zero = {0.f, 0.f, 0.f, 0.f, 0.f, 0.f, 0.f, 0.f};
    for (int t = 0; t < tilesX; ++t) {
        const float* ap = Xb + ((size_t)(t * 16 + col) * 4 + 2 * hi);
        v2f a;
        a.x = ap[0];
        a.y = ap[1];
        v8f acc = __builtin_amdgcn_wmma_f32_16x16x4_f32(
            false, a, false, bb, (short)0, zero, false, false);
        float m = acc[0];
#pragma unroll
        for (int v = 1; v < 8; ++v) m = fminf(m, acc[v]);
        // combine rows 0-7 (lane j) with rows 8-15 (lane j+16): same column
        m = fminf(m, __shfl_xor(m, 16, 32));
        runCol = fminf(runCol, m + yn);
    }

    // lanes 0-15 hold final column mins (16-31 are duplicates); sum within group
    float s = runCol;
    s += __shfl_xor(s, 1, 32);
    s += __shfl_xor(s, 2, 32);
    s += __shfl_xor(s, 4, 32);
    s += __shfl_xor(s, 8, 32);
    if (lane == 0) partial[wave] = s;
}

// ---------------------------------------------------------------------------
// Deterministic final reduction (single block, fixed tree).
// ---------------------------------------------------------------------------
__device__ float block_sum(const float* __restrict__ p, int n, float* sh) {
    float s = 0.0f;
    for (int i = (int)threadIdx.x; i < n; i += (int)blockDim.x) s += p[i];
    sh[threadIdx.x] = s;
    __syncthreads();
    for (int off = 128; off > 0; off >>= 1) {
        if ((int)threadIdx.x < off) sh[threadIdx.x] += sh[threadIdx.x + off];
        __syncthreads();
    }
    float r = sh[0];
    __syncthreads();
    return r;
}

__global__ void finalize_kernel(const float* __restrict__ pRF, int nRF,
                                const float* __restrict__ pCF, int nCF,
                                const float* __restrict__ pRC, int nRC,
                                const float* __restrict__ pCC, int nCC,
                                const float* __restrict__ alpha,
                                float* __restrict__ out) {
    __shared__ float sh[256];
    float sRF = block_sum(pRF, nRF, sh);   // sum of fine row-mins
    float sCF = block_sum(pCF, nCF, sh);   // sum of fine col-mins
    float sRC = block_sum(pRC, nRC, sh);   // sum of coarse row-mins
    float sCC = block_sum(pCC, nCC, sh);   // sum of coarse col-mins
    if (threadIdx.x == 0) {
        float loss_fine   = sRF / (float)(BATCH * N_FINE)   + sCF / (float)(BATCH * N_GT);
        float loss_coarse = sRC / (float)(BATCH * N_COARSE) + sCC / (float)(BATCH * N_GT);
        float loss = loss_coarse + alpha[0] * loss_fine;
        out[0] = loss;
        out[1] = loss_coarse;
        out[2] = loss_fine;
    }
}

// ---------------------------------------------------------------------------
// Host launch
// ---------------------------------------------------------------------------
extern "C" void kernel_launch(void* const* d_in, const int* in_sizes, int n_in,
                              void* d_out, int out_size, void* d_ws, size_t ws_size,
                              hipStream_t stream) {
    const float* coarse = (const float*)d_in[0];   // [B, 1024, 3]
    const float* fine   = (const float*)d_in[1];   // [B, 8192, 3]
    const float* gt     = (const float*)d_in[2];   // [B, 3, 8192]
    const float* alpha  = (const float*)d_in[3];   // scalar
    float* out = (float*)d_out;

    // workspace layout (floats)
    float* ws = (float*)d_ws;
    float* X4F = ws;                                   // 8*8192*4 = 262144
    float* X4C = X4F + (size_t)BATCH * N_FINE * 4;     // 8*1024*4 =  32768
    float* Y4  = X4C + (size_t)BATCH * N_COARSE * 4;   // 8*8192*4 = 262144
    float* pRF = Y4  + (size_t)BATCH * N_GT * 4;       // 4096
    float* pCF = pRF + (size_t)BATCH * TILES_F;        // 4096
    float* pRC = pCF + (size_t)BATCH * TILES_G;        //  512
    float* pCC = pRC + (size_t)BATCH * TILES_C;        // 4096

    const int nPF = BATCH * N_FINE;    // 65536
    const int nPC = BATCH * N_COARSE;  //  8192
    const int nPG = BATCH * N_GT;      // 65536

    prep_x_kernel<<<(nPF + 255) / 256, 256, 0, stream>>>(fine, X4F, nPF);
    prep_x_kernel<<<(nPC + 255) / 256, 256, 0, stream>>>(coarse, X4C, nPC);
    prep_y_kernel<<<(nPG + 255) / 256, 256, 0, stream>>>(gt, Y4, N_GT, nPG);

    // fine chamfer: 8*512 waves each direction (256 thr = 8 waves/block)
    chamfer_rows_kernel<<<(BATCH * TILES_F) / 8, 256, 0, stream>>>(
        X4F, Y4, TILES_F, TILES_G, N_FINE, N_GT, pRF);
    chamfer_cols_kernel<<<(BATCH * TILES_G) / 8, 256, 0, stream>>>(
        X4F, Y4, TILES_F, TILES_G, N_FINE, N_GT, pCF);

    // coarse chamfer
    chamfer_rows_kernel<<<(BATCH * TILES_C) / 8, 256, 0, stream>>>(
        X4C, Y4, TILES_C, TILES_G, N_COARSE, N_GT, pRC);
    chamfer_cols_kernel<<<(BATCH * TILES_G) / 8, 256, 0, stream>>>(
        X4C, Y4, TILES_C, TILES_G, N_COARSE, N_GT, pCC);

    finalize_kernel<<<1, 256, 0, stream>>>(
        pRF, BATCH * TILES_F, pCF, BATCH * TILES_G,
        pRC, BATCH * TILES_C, pCC, BATCH * TILES_G,
        alpha, out);
}